// GtNet_70531952935098
// MI455X (gfx1250) — compile-verified
//
#include <hip/hip_runtime.h>
#include <stdint.h>

// Problem constants (fixed by the reference)
#define HH 256
#define WW 256
#define BB 8
#define HW (HH * WW)

#define TILE 16
#define HALO 20               // TILE + 2*M_RANGE
#define PLANE (HALO * HALO)   // 400 elements per staged plane

// gfx1250 async global->LDS dword copy (ASYNCcnt-tracked, EXEC-masked).
// vdst = LDS byte address VGPR, vaddr = signed 32-bit byte offset VGPR,
// saddr = 64-bit SGPR base (GVS mode).
__device__ __forceinline__ void async_copy_dword(uint32_t lds_addr,
                                                 const void* sbase,
                                                 uint32_t byte_off) {
  asm volatile("global_load_async_to_lds_b32 %0, %1, %2"
               :: "v"(lds_addr), "v"(byte_off), "s"(sbase)
               : "memory");
}

__device__ __forceinline__ void wait_asynccnt0() {
  asm volatile("s_wait_asynccnt 0" ::: "memory");
}

// Bilinear "hat" weight: identical to the reference's (1-f)/f spread.
// For bin k (0..4) and motion m in [-2,2): w = max(0, 1 - |m - (k-2)|).
__device__ __forceinline__ float hatw(float m, int k) {
  return fmaxf(1.0f - fabsf(m - (float)(k - 2)), 0.0f);
}

__global__ __launch_bounds__(256) void gtnet_kernel(
    const float* __restrict__ im_input,   // (B,6,H,W) — use channels 3..5
    const float* __restrict__ gt_motion,  // (B,2,H,W)
    const int*   __restrict__ gt_depth,   // (B,1,H,W), values {0,1}
    float* __restrict__ out)              // pred | m_mask | d_mask | 1-seg
{
  // Planes: 0=motion.x  1=motion.y  2=depth  3..5=image ch 3..5
  __shared__ uint32_t smem[6 * PLANE];

  const int lx = threadIdx.x, ly = threadIdx.y;
  const int t  = ly * TILE + lx;
  const int x0 = blockIdx.x * TILE;
  const int y0 = blockIdx.y * TILE;
  const int b  = blockIdx.z;

  const float* mx_base = gt_motion + (size_t)(b * 2 + 0) * HW;
  const float* my_base = gt_motion + (size_t)(b * 2 + 1) * HW;
  const int*   dp_base = gt_depth  + (size_t)b * HW;
  const float* i0_base = im_input  + (size_t)(b * 6 + 3) * HW;
  const float* i1_base = im_input  + (size_t)(b * 6 + 4) * HW;
  const float* i2_base = im_input  + (size_t)(b * 6 + 5) * HW;

  // ---- Stage 20x20 halo of all 6 planes into LDS via async copies ----
  #pragma unroll
  for (int h0 = 0; h0 < PLANE; h0 += 256) {
    const int h = h0 + t;
    if (h < PLANE) {
      const int hy = h / HALO, hx = h % HALO;
      const int gy = y0 + hy - 2, gx = x0 + hx - 2;
      const bool valid = (gy >= 0) & (gy < HH) & (gx >= 0) & (gx < WW);
      if (valid) {
        const uint32_t off = (uint32_t)(gy * WW + gx) * 4u;
        async_copy_dword((uint32_t)(uintptr_t)&smem[0 * PLANE + h], mx_base, off);
        async_copy_dword((uint32_t)(uintptr_t)&smem[1 * PLANE + h], my_base, off);
        async_copy_dword((uint32_t)(uintptr_t)&smem[2 * PLANE + h], dp_base, off);
        async_copy_dword((uint32_t)(uintptr_t)&smem[3 * PLANE + h], i0_base, off);
        async_copy_dword((uint32_t)(uintptr_t)&smem[4 * PLANE + h], i1_base, off);
        async_copy_dword((uint32_t)(uintptr_t)&smem[5 * PLANE + h], i2_base, off);
      } else {
        smem[0 * PLANE + h] = 0u;           // motion -> 0 (never used alone)
        smem[1 * PLANE + h] = 0u;
        smem[2 * PLANE + h] = 0xFFFFFFFFu;  // depth sentinel: neither layer
        smem[3 * PLANE + h] = 0u;           // image zero-pad
        smem[4 * PLANE + h] = 0u;
        smem[5 * PLANE + h] = 0u;
      }
    }
  }
  wait_asynccnt0();   // each wave drains its own ASYNCcnt
  __syncthreads();    // whole workgroup syncs (also covers the LDS fills)

  // ---- Own-pixel quantities: m_mask weights + depth one-hot ----
  const int cix = (ly + 2) * HALO + (lx + 2);
  const float    mxp = __uint_as_float(smem[0 * PLANE + cix]);
  const float    myp = __uint_as_float(smem[1 * PLANE + cix]);
  const uint32_t dpp = smem[2 * PLANE + cix];

  float wx[5], wy[5];
  #pragma unroll
  for (int k = 0; k < 5; ++k) {
    wx[k] = hatw(mxp, k);
    wy[k] = hatw(myp, k);
  }

  // ---- Pass 1: flow + per-layer probability totals over 25 classes ----
  float flow[25];
  float total1 = 0.0f, total0 = 0.0f;
  #pragma unroll
  for (int cy = 0; cy < 5; ++cy) {
    #pragma unroll
    for (int cx = 0; cx < 5; ++cx) {
      const int c   = cy * 5 + cx;
      const int pos = (ly + cy) * HALO + (lx + cx);   // (+2 halo) + (shift-2)
      const float    ms = __uint_as_float(smem[0 * PLANE + pos]);
      const float    mt = __uint_as_float(smem[1 * PLANE + pos]);
      const uint32_t d  = smem[2 * PLANE + pos];
      const float fl = hatw(ms, cx) * hatw(mt, cy);
      flow[c] = fl;
      total1 += (d == 1u) ? fl : 0.0f;
      total0 += (d == 0u) ? fl : 0.0f;
    }
  }
  // Hoisted epsilons: reference adds 1e-8 per class before summing.
  total1 += 25.0f * 1e-8f;
  total0 += 25.0f * 1e-8f;

  // Two-layer compositing scales (layer 1 first, then layer 0)
  const float scale1 = fminf(1.0f / total1, 1.0f);          // left = 1
  const float sum1   = total1 * scale1;
  const float left   = fmaxf(1.0f - sum1, 0.0f);
  const float scale0 = fminf(left / total0, 1.0f);
  const float seg    = sum1 + total0 * scale0;
  const float epsterm = 1e-8f * (scale1 + scale0);

  // ---- Pass 2: pred = sum_c im(shifted) * out_mask[c] ----
  float pr0 = 0.0f, pr1 = 0.0f, pr2 = 0.0f;
  #pragma unroll
  for (int cy = 0; cy < 5; ++cy) {
    #pragma unroll
    for (int cx = 0; cx < 5; ++cx) {
      const int c   = cy * 5 + cx;
      const int pos = (ly + cy) * HALO + (lx + cx);
      const uint32_t d = smem[2 * PLANE + pos];
      const float sel  = (d == 1u) ? scale1 : ((d == 0u) ? scale0 : 0.0f);
      const float oc   = fmaf(flow[c], sel, epsterm);
      pr0 = fmaf(__uint_as_float(smem[3 * PLANE + pos]), oc, pr0);
      pr1 = fmaf(__uint_as_float(smem[4 * PLANE + pos]), oc, pr1);
      pr2 = fmaf(__uint_as_float(smem[5 * PLANE + pos]), oc, pr2);
    }
  }

  // ---- Write outputs (flat, in reference return order) ----
  const int y = y0 + ly, x = x0 + lx;
  const size_t pix   = (size_t)y * WW + x;
  const size_t MMASK = (size_t)BB * 3 * HW;            // after pred
  const size_t DMASK = MMASK + (size_t)BB * 25 * HW;   // after m_mask
  const size_t SEGB  = DMASK + (size_t)BB * 2 * HW;    // after d_mask

  out[(size_t)(b * 3 + 0) * HW + pix] = pr0;
  out[(size_t)(b * 3 + 1) * HW + pix] = pr1;
  out[(size_t)(b * 3 + 2) * HW + pix] = pr2;

  #pragma unroll
  for (int cy = 0; cy < 5; ++cy)
    #pragma unroll
    for (int cx = 0; cx < 5; ++cx)
      out[MMASK + (size_t)(b * 25 + cy * 5 + cx) * HW + pix] = wy[cy] * wx[cx];

  out[DMASK + (size_t)(b * 2 + 0) * HW + pix] = (dpp == 0u) ? 1.0f : 0.0f;
  out[DMASK + (size_t)(b * 2 + 1) * HW + pix] = (dpp == 1u) ? 1.0f : 0.0f;

  out[SEGB + (size_t)b * HW + pix] = 1.0f - seg;
}

extern "C" void kernel_launch(void* const* d_in, const int* in_sizes, int n_in,
                              void* d_out, int out_size, void* d_ws, size_t ws_size,
                              hipStream_t stream) {
  (void)in_sizes; (void)n_in; (void)out_size; (void)d_ws; (void)ws_size;
  const float* im_input  = (const float*)d_in[0];
  const float* gt_motion = (const float*)d_in[1];
  const int*   gt_depth  = (const int*)d_in[2];
  // d_in[3] (m_kernel) is a fixed one-hot shift bank; its structure is
  // hardcoded as the (cy-2, cx-2) shifts above.

  dim3 block(TILE, TILE, 1);
  dim3 grid(WW / TILE, HH / TILE, BB);
  gtnet_kernel<<<grid, block, 0, stream>>>(im_input, gt_motion, gt_depth,
                                           (float*)d_out);
}